// MultiBoxLoss_20744692040220
// MI455X (gfx1250) — compile-verified
//
#include <hip/hip_runtime.h>
#include <hip/hip_bf16.h>

// ---------------- problem constants (match reference) ----------------
#define BB 64          // batch (num gt boxes)
#define AA 65536       // anchors
#define TH_HIGH 0.6f
#define TH_LOW  0.3f
#define VAR0 0.1f
#define VAR1 0.2f
#define IMG_SIZE 255.0f
#define BDIM 256
#define NB1 (BB * (AA / BDIM))   // 16384 pass-1 blocks

typedef __attribute__((ext_vector_type(2))) float v2f;
typedef __attribute__((ext_vector_type(8))) float v8f;
typedef __attribute__((ext_vector_type(4))) int   v4i;

#define HAS_WMMA_F32_4 __has_builtin(__builtin_amdgcn_wmma_f32_16x16x4_f32)
#define HAS_ASYNC_LDS  __has_builtin(__builtin_amdgcn_global_load_async_to_lds_b128)

#if HAS_ASYNC_LDS
typedef __attribute__((address_space(1))) v4i* glob_v4i_ptr;
typedef __attribute__((address_space(3))) v4i* lds_v4i_ptr;
#endif

// ---------------- helpers ----------------
__device__ __forceinline__ float smooth_l1(float x) {
    float ax = fabsf(x);
    return (ax < 1.0f) ? 0.5f * x * x : ax - 0.5f;
}

__device__ __forceinline__ float blockReduceF(float v, float* red) {
    int tid = threadIdx.x;
    red[tid] = v;
    __syncthreads();
    for (int s = BDIM / 2; s > 0; s >>= 1) {
        if (tid < s) red[tid] += red[tid + s];
        __syncthreads();
    }
    float r = red[0];
    __syncthreads();
    return r;
}

// ---------------- init: zero per-row positive counts ----------------
__global__ void mb_init(int* num_pos) {
    if (threadIdx.x < BB) num_pos[threadIdx.x] = 0;
}

// ---------------- pass 1: fused streaming pass ----------------
// IoU match, encode + smooth-L1, log-softmax NLL, decode diagnostics.
// Writes nll[B,A], flags[B,A] (bit0=pos, bit1=label0, bit2=pred1),
// per-block partials (4 quantities), per-row positive counts.
__global__ __launch_bounds__(BDIM) void mb_pass1(
    const float* __restrict__ loc, const float* __restrict__ conf,
    const float* __restrict__ gt, const float* __restrict__ anchors,
    float* __restrict__ nll_out, unsigned char* __restrict__ flags,
    float* __restrict__ partials1, int* __restrict__ num_pos) {
    __shared__ float lds_anchor[BDIM * 4];
    __shared__ float red[BDIM];

    const int tid = threadIdx.x;
    const int b  = blockIdx.y;
    const int a0 = blockIdx.x * BDIM;
    const int a  = a0 + tid;

    // Stage this block's anchor tile (4 KiB) into LDS via CDNA5 async copy.
#if HAS_ASYNC_LDS
    __builtin_amdgcn_global_load_async_to_lds_b128(
        (glob_v4i_ptr)(anchors + (size_t)a * 4),
        (lds_v4i_ptr)&lds_anchor[tid * 4], 0, 0);
#if __has_builtin(__builtin_amdgcn_s_wait_asynccnt)
    __builtin_amdgcn_s_wait_asynccnt(0);
#else
    asm volatile("s_wait_asynccnt 0" ::: "memory");
#endif
#else
    {
        float4 av = ((const float4*)anchors)[a];
        lds_anchor[tid * 4 + 0] = av.x;
        lds_anchor[tid * 4 + 1] = av.y;
        lds_anchor[tid * 4 + 2] = av.z;
        lds_anchor[tid * 4 + 3] = av.w;
    }
#endif
    __syncthreads();

    // GT box (point form)
    const float gx0 = gt[b * 4 + 0], gy0 = gt[b * 4 + 1];
    const float gx1 = gt[b * 4 + 2], gy1 = gt[b * 4 + 3];

    // Anchor (center-size) + point form
    const float acx = lds_anchor[tid * 4 + 0], acy = lds_anchor[tid * 4 + 1];
    const float aw  = lds_anchor[tid * 4 + 2], ah  = lds_anchor[tid * 4 + 3];
    const float ax0 = acx - 0.5f * aw, ay0 = acy - 0.5f * ah;
    const float ax1 = acx + 0.5f * aw, ay1 = acy + 0.5f * ah;

    // IoU
    const float ltx = fmaxf(gx0, ax0), lty = fmaxf(gy0, ay0);
    const float rbx = fminf(gx1, ax1), rby = fminf(gy1, ay1);
    const float iw = fmaxf(rbx - ltx, 0.0f), ih = fmaxf(rby - lty, 0.0f);
    const float inter = iw * ih;
    const float areaA = (gx1 - gx0) * (gy1 - gy0);
    const float areaB = (ax1 - ax0) * (ay1 - ay0);
    const float iou = inter / (areaA + areaB - inter);

    const bool pos  = (iou >= TH_HIGH);
    const bool lbl0 = (iou <= TH_LOW);    // labels == 0

    // Predictions
    const float4 lp = ((const float4*)loc)[(size_t)b * AA + a];
    const float2 cp = ((const float2*)conf)[(size_t)b * AA + a];

    // NLL of target class under log-softmax
    const float m   = fmaxf(cp.x, cp.y);
    const float lse = m + logf(expf(cp.x - m) + expf(cp.y - m));
    const float nll = lse - (pos ? cp.y : cp.x);
    const bool pred1 = (cp.y > cp.x);

    // encode(gt, anchor) + smooth L1 vs loc_pred
    const float mcx = 0.5f * (gx0 + gx1), mcy = 0.5f * (gy0 + gy1);
    const float mw = gx1 - gx0, mh = gy1 - gy0;
    const float e0 = (mcx - acx) / (VAR0 * aw);
    const float e1 = (mcy - acy) / (VAR0 * ah);
    const float e2 = logf(mw / aw) / VAR1;
    const float e3 = logf(mh / ah) / VAR1;
    const float sl = smooth_l1(lp.x - e0) + smooth_l1(lp.y - e1) +
                     smooth_l1(lp.z - e2) + smooth_l1(lp.w - e3);

    // decode diagnostics
    const float dcx = acx + lp.x * VAR0 * aw;
    const float dcy = acy + lp.y * VAR0 * ah;
    const float dw = aw * expf(lp.z * VAR1);
    const float dh = ah * expf(lp.w * VAR1);
    const float dx0 = dcx - 0.5f * dw, dy0 = dcy - 0.5f * dh;
    const float dx1 = dx0 + dw, dy1 = dy0 + dh;
    const float px = (gx0 - dx0) * IMG_SIZE, py = (gy0 - dy0) * IMG_SIZE;
    const float perr = sqrtf(px * px + py * py);
    const float sdx = fabsf(gx1 - dx1), sdy = fabsf(gy1 - dy1);

    const size_t gi = (size_t)b * AA + a;
    nll_out[gi] = nll;
    flags[gi] = (unsigned char)((pos ? 1 : 0) | (lbl0 ? 2 : 0) | (pred1 ? 4 : 0));

    const float pf = pos ? 1.0f : 0.0f;
    const float s0 = blockReduceF(pf * sl,   red);
    const float s1 = blockReduceF(pf * perr, red);
    const float s2 = blockReduceF(pf * sdx,  red);
    const float s3 = blockReduceF(pf * sdy,  red);
    const float sc = blockReduceF(pf,        red);
    if (tid == 0) {
        const int nb = blockIdx.y * gridDim.x + blockIdx.x;
        partials1[0 * NB1 + nb] = s0;
        partials1[1 * NB1 + nb] = s1;
        partials1[2 * NB1 + nb] = s2;
        partials1[3 * NB1 + nb] = s3;
        atomicAdd(&num_pos[b], (int)(sc + 0.5f));  // exact integer sum
    }
}

// ---------------- pass 2: per-row radix select (k-th largest CE) ----------------
// ce = lbl0 ? nll : 0 ; nll >= 0 so IEEE bits are order-preserving.
__global__ __launch_bounds__(BDIM) void mb_pass2(
    const float* __restrict__ nll, const unsigned char* __restrict__ flags,
    const int* __restrict__ num_pos,
    unsigned int* __restrict__ Tkey, int* __restrict__ Rrem) {
    __shared__ int hist[256];
    __shared__ unsigned int s_prefix;
    __shared__ int s_remaining;

    const int b = blockIdx.x, tid = threadIdx.x;
    int k = 3 * num_pos[b];
    k = max(k, 10);
    k = min(k, AA - 1);

    unsigned int prefix = 0;
    int remaining = k;

    for (int level = 0; level < 4; ++level) {
        hist[tid] = 0;
        __syncthreads();
        const int sh = 24 - 8 * level;
        for (int a = tid; a < AA; a += BDIM) {
            const size_t gi = (size_t)b * AA + a;
            const unsigned char f = flags[gi];
            const unsigned int key = (f & 2) ? __float_as_uint(nll[gi]) : 0u;
            const bool match = (level == 0) || ((key >> (sh + 8)) == prefix);
            if (match) atomicAdd(&hist[(key >> sh) & 0xFF], 1);
        }
        __syncthreads();
        if (tid == 0) {
            int acc = 0, v;
            for (v = 255; v >= 1; --v) {
                if (acc + hist[v] >= remaining) break;
                acc += hist[v];
            }
            s_prefix = (prefix << 8) | (unsigned int)v;
            s_remaining = remaining - acc;   // tie slots still to fill
        }
        __syncthreads();
        prefix = s_prefix;
        remaining = s_remaining;
        __syncthreads();
    }
    if (tid == 0) {
        Tkey[b] = prefix;     // exact k-th largest key
        Rrem[b] = remaining;  // how many ties at Tkey are selected (by index order)
    }
}

// ---------------- pass 3: stable tie-break scan + classification stats ----------------
__global__ __launch_bounds__(BDIM) void mb_pass3(
    const float* __restrict__ nll, const unsigned char* __restrict__ flags,
    const unsigned int* __restrict__ Tkey, const int* __restrict__ Rrem,
    float* __restrict__ partials3) {
    __shared__ int scan[BDIM];
    __shared__ int s_base;
    __shared__ float red[BDIM];

    const int b = blockIdx.x, tid = threadIdx.x;
    const unsigned int T = Tkey[b];
    const int r = Rrem[b];
    if (tid == 0) s_base = 0;
    __syncthreads();

    float cls_num = 0, wsum = 0, posCor = 0, posTot = 0, negCor = 0, negTot = 0;

    for (int a0 = 0; a0 < AA; a0 += BDIM) {
        const size_t gi = (size_t)b * AA + a0 + tid;
        const unsigned char f = flags[gi];
        const float nv = nll[gi];
        const unsigned int key = (f & 2) ? __float_as_uint(nv) : 0u;
        const bool pos = f & 1, pred1 = f & 4;
        const int isTie = (key == T) ? 1 : 0;

        // inclusive scan of tie flags (index order within chunk)
        scan[tid] = isTie;
        __syncthreads();
        for (int off = 1; off < BDIM; off <<= 1) {
            int v = (tid >= off) ? scan[tid - off] : 0;
            __syncthreads();
            scan[tid] += v;
            __syncthreads();
        }
        const int excl = scan[tid] - isTie;
        const int chunkTotal = scan[BDIM - 1];
        const int base = s_base;
        __syncthreads();
        if (tid == 0) s_base = base + chunkTotal;

        const bool neg = (key > T) || (isTie && (base + excl) < r);
        const bool sel = pos || neg;
        const float w = sel ? (pos ? 1.0f : (1.0f / 3.0f)) : 0.0f;
        cls_num += nv * w;
        wsum += w;
        if (pos) { posTot += 1.0f; if (pred1) posCor += 1.0f; }
        else if (neg) { negTot += 1.0f; if (!pred1) negCor += 1.0f; }
        __syncthreads();
    }

    const float q0 = blockReduceF(cls_num, red);
    const float q1 = blockReduceF(wsum,    red);
    const float q2 = blockReduceF(posCor,  red);
    const float q3 = blockReduceF(posTot,  red);
    const float q4 = blockReduceF(negCor,  red);
    const float q5 = blockReduceF(negTot,  red);
    if (tid == 0) {
        partials3[0 * BB + b] = q0;
        partials3[1 * BB + b] = q1;
        partials3[2 * BB + b] = q2;
        partials3[3 * BB + b] = q3;
        partials3[4 * BB + b] = q4;
        partials3[5 * BB + b] = q5;
    }
}

// ---------------- pass 4: WMMA ones-matrix reduction of partial sums ----------------
// D = ones(16x4) x B(4x16) + C accumulates 64 values per issue into 16 column
// sums; final total = sum_n D[0,n] (VGPR0, lanes 0..15). One wave per quantity.
__global__ __launch_bounds__(32) void mb_reduce(
    const float* __restrict__ partials1, const float* __restrict__ partials3,
    float* __restrict__ sums) {
    const int q = blockIdx.x;
    const float* src;
    int cnt;
    if (q < 4) { src = partials1 + q * NB1; cnt = NB1; }
    else       { src = partials3 + (q - 4) * BB; cnt = BB; }
    const int lane = threadIdx.x;

#if HAS_WMMA_F32_4
    v2f a; a[0] = 1.0f; a[1] = 1.0f;           // A = ones(16x4)
    v8f c = {0, 0, 0, 0, 0, 0, 0, 0};
    for (int i = 0; i < cnt; i += 64) {
        v2f bv;
        bv[0] = src[i + lane];
        bv[1] = src[i + 32 + lane];
        c = __builtin_amdgcn_wmma_f32_16x16x4_f32(
            false, a, false, bv, (short)0, c, false, false);
    }
    __shared__ float red[16];
    if (lane < 16) red[lane] = c[0];           // D row 0 = VGPR0, lanes 0..15
    __syncthreads();
    if (lane == 0) {
        float s = 0;
        for (int n = 0; n < 16; ++n) s += red[n];
        sums[q] = s;
    }
#else
    float acc = 0;
    for (int i = lane; i < cnt; i += 32) acc += src[i];
    __shared__ float red[32];
    red[lane] = acc;
    __syncthreads();
    if (lane == 0) {
        float s = 0;
        for (int n = 0; n < 32; ++n) s += red[n];
        sums[q] = s;
    }
#endif
}

// ---------------- pass 5: final scalars ----------------
__global__ void mb_final(const float* __restrict__ sums,
                         const int* __restrict__ num_pos,
                         float* __restrict__ out) {
    int N = 0;
    for (int b = 0; b < BB; ++b) N += num_pos[b];
    const float Nf = fmaxf((float)N, 1.0f);
    out[0] = sums[0] / (Nf * 4.0f);                 // loss_loc
    out[1] = sums[4] / sums[5];                     // loss_cls
    out[2] = sums[6] / fmaxf(sums[7], 1.0f);        // pos_accuracy
    out[3] = sums[8] / fmaxf(sums[9], 1.0f);        // neg_accuracy
    out[4] = sums[1] / Nf;                          // position_error
    out[5] = sums[2] / Nf * IMG_SIZE;               // size_errors[0]
    out[6] = sums[3] / Nf * IMG_SIZE;               // size_errors[1]
    out[7] = (float)N;                              // N
}

// ---------------- launch ----------------
extern "C" void kernel_launch(void* const* d_in, const int* in_sizes, int n_in,
                              void* d_out, int out_size, void* d_ws, size_t ws_size,
                              hipStream_t stream) {
    const float* loc     = (const float*)d_in[0];  // [B,A,4]
    const float* conf    = (const float*)d_in[1];  // [B,A,2]
    const float* gt      = (const float*)d_in[2];  // [B,4]
    const float* anchors = (const float*)d_in[3];  // [A,4]
    float* out = (float*)d_out;

    // Deterministic workspace carve-out (~20.5 MiB)
    float* nll = (float*)d_ws;                                   // B*A floats
    unsigned char* flags = (unsigned char*)(nll + (size_t)BB * AA); // B*A bytes
    float* partials1 = (float*)(flags + (size_t)BB * AA);        // 4 * NB1
    float* partials3 = partials1 + 4 * NB1;                      // 6 * BB
    int* num_pos = (int*)(partials3 + 6 * BB);                   // BB
    unsigned int* Tkey = (unsigned int*)(num_pos + BB);          // BB
    int* Rrem = (int*)(Tkey + BB);                               // BB
    float* sums = (float*)(Rrem + BB);                           // 16

    mb_init<<<1, 64, 0, stream>>>(num_pos);
    mb_pass1<<<dim3(AA / BDIM, BB), BDIM, 0, stream>>>(
        loc, conf, gt, anchors, nll, flags, partials1, num_pos);
    mb_pass2<<<BB, BDIM, 0, stream>>>(nll, flags, num_pos, Tkey, Rrem);
    mb_pass3<<<BB, BDIM, 0, stream>>>(nll, flags, Tkey, Rrem, partials3);
    mb_reduce<<<10, 32, 0, stream>>>(partials1, partials3, sums);
    mb_final<<<1, 1, 0, stream>>>(sums, num_pos, out);
}